// PoolProd_6871947674133
// MI455X (gfx1250) — compile-verified
//
#include <hip/hip_runtime.h>
#include <stdint.h>

// segment_prod(feats[2M,64] fp32, sorted batch[2M] -> 8192 segments).
// Memory-bound: ~520 MB traffic -> ~22 us floor at 23.3 TB/s HBM.
// One wave32 workgroup per segment (binary search is pure SALU/SMEM).
// CDNA5 async global->LDS double buffering (GLOBAL_LOAD_ASYNC_TO_LDS_B128
// + S_WAIT_ASYNCCNT) streams 8 KB chunks while the previous chunk's
// multiply chain runs; single-wave workgroups need no barriers.
// Async INST_OFFSET is added to BOTH the LDS and global addresses, so all
// 16 issues of a chunk share one VGPR address pair + immediate offsets.

#define D_FEAT 64
#define ROWS_PER_CHUNK 32
#define CHUNK_FLOATS (ROWS_PER_CHUNK * D_FEAT)     // 2048 floats = 8 KB
#define CHUNK_BYTES  (CHUNK_FLOATS * 4)
#define CHUNK_VEC4   (CHUNK_FLOATS / 4)            // 512 x 16B transfers
#define ISSUES_PER_CHUNK (CHUNK_VEC4 / 32)         // 16 async b128 per lane

__device__ __forceinline__ uint32_t lds_offset(const void* p) {
    // Generic pointers into LDS carry the wave-relative LDS byte offset in
    // their low 32 bits (ISA 10.2).
    return (uint32_t)(uintptr_t)p;
}

__global__ __launch_bounds__(32) void PoolProd_segprod_kernel(
    const float* __restrict__ feats,
    const int*   __restrict__ batch,
    float*       __restrict__ out,
    int nrows)
{
    __shared__ float buf[2][CHUNK_FLOATS];

    const int seg  = (int)blockIdx.x;
    const int lane = (int)threadIdx.x;

    // lower_bound(seg) / lower_bound(seg+1) over sorted batch (uniform ->
    // compiler emits SALU loop with scalar s_load, zero VMEM pressure).
    int lo = 0, hi = nrows;
    while (lo < hi) { int mid = (lo + hi) >> 1; if (batch[mid] < seg)     lo = mid + 1; else hi = mid; }
    const int start = lo;
    hi = nrows;
    while (lo < hi) { int mid = (lo + hi) >> 1; if (batch[mid] < seg + 1) lo = mid + 1; else hi = mid; }
    const int end = lo;

    const int total = end - start;
    const int nch   = (total + ROWS_PER_CHUNK - 1) / ROWS_PER_CHUNK;

    const uint64_t gbase =
        (uint64_t)(uintptr_t)feats + (uint64_t)start * (D_FEAT * 4) +
        (uint64_t)(lane * 16);

    // Issue one 32-row chunk (8 KB) into buf[par]: 16 async b128 per lane.
    // One address pair per chunk; issues differ only by immediate offset,
    // which hardware applies to both the LDS and the global address.
    auto issue_chunk = [&](int c, int par) {
        const uint64_t g0 = gbase + (uint64_t)c * CHUNK_BYTES;
        const uint32_t l0 = lds_offset(&buf[par][0]) + (uint32_t)(lane * 16);
        const int rows = min(ROWS_PER_CHUNK, total - c * ROWS_PER_CHUNK);
        if (rows == ROWS_PER_CHUNK) {
            // Full chunk: unguarded steady-state issues.
#pragma unroll
            for (int i = 0; i < ISSUES_PER_CHUNK; ++i) {
                asm volatile("global_load_async_to_lds_b128 %0, %1, off offset:%c2"
                             :: "v"(l0), "v"(g0), "i"(i * 512) : "memory");
            }
        } else {
            // Tail chunk: EXEC-predicate out-of-range lanes.
            const int limit = rows * (D_FEAT / 4);
#pragma unroll
            for (int i = 0; i < ISSUES_PER_CHUNK; ++i) {
                if (i * 32 + lane < limit) {
                    asm volatile("global_load_async_to_lds_b128 %0, %1, off offset:%c2"
                                 :: "v"(l0), "v"(g0), "i"(i * 512) : "memory");
                }
            }
        }
    };

    float accx = 1.0f, accy = 1.0f;                  // lane owns cols 2t, 2t+1

    if (nch > 0) issue_chunk(0, 0);

    for (int c = 0; c < nch; ++c) {
        const int par = c & 1;
        // Chunk c's async transfers landed in buf[par].
        asm volatile("s_wait_asynccnt 0x0" ::: "memory");
        if (c + 1 < nch) {
            // WAR fence: prior DS reads of buf[par^1] retire before the
            // async engine may overwrite that buffer.
            asm volatile("s_wait_dscnt 0x0" ::: "memory");
            issue_chunk(c + 1, par ^ 1);             // overlaps compute below
        }
        const int rows = min(ROWS_PER_CHUNK, total - c * ROWS_PER_CHUNK);
        const float* b = &buf[par][0];
#pragma unroll 8
        for (int r = 0; r < rows; ++r) {
            // Contiguous pair -> ds_load_b64; banks 2t,2t+1: conflict-free.
            accx *= b[r * D_FEAT + 2 * lane];
            accy *= b[r * D_FEAT + 2 * lane + 1];
        }
    }

    // Empty segments fall through with acc == 1.0 (multiplicative identity).
    float2* o = (float2*)(out + (size_t)seg * D_FEAT);
    o[lane] = make_float2(accx, accy);
}

extern "C" void kernel_launch(void* const* d_in, const int* in_sizes, int n_in,
                              void* d_out, int out_size, void* d_ws, size_t ws_size,
                              hipStream_t stream) {
    (void)n_in; (void)d_ws; (void)ws_size;
    const float* feats = (const float*)d_in[0];
    const int*   batch = (const int*)d_in[1];
    float*       out   = (float*)d_out;

    const int nrows = in_sizes[1];          // 2,000,000
    const int nseg  = out_size / D_FEAT;    // 8192

    PoolProd_segprod_kernel<<<dim3((unsigned)nseg), dim3(32), 0, stream>>>(
        feats, batch, out, nrows);
}